// GCN_50654844289592
// MI455X (gfx1250) — compile-verified
//
#include <hip/hip_runtime.h>

#define N_NODES 100000
#define N_EDGES 1600000
#define DIM 128           // IN_DIM == HIDDEN == 128
#define TILES_N2 (DIM / 32)   // 32-wide column tiles per wave

typedef float v2f __attribute__((ext_vector_type(2)));
typedef float v8f __attribute__((ext_vector_type(8)));

// ---------------- degree / normalization ----------------

__global__ void k_deg_init(float* __restrict__ deg) {
  int i = blockIdx.x * blockDim.x + threadIdx.x;
  if (i < N_NODES) deg[i] = 1.0f;   // self-loop
}

__global__ void k_deg_count(const int* __restrict__ col, float* __restrict__ deg) {
  int i = blockIdx.x * blockDim.x + threadIdx.x;
  if (i < N_EDGES) atomicAdd(&deg[col[i]], 1.0f);
}

__global__ void k_dinv(float* __restrict__ deg) {
  int i = blockIdx.x * blockDim.x + threadIdx.x;
  if (i < N_NODES) deg[i] = rsqrtf(deg[i]);   // in place: deg -> dinv
}

// ---------------- dense GEMM via f32 WMMA ----------------
// H[N,128] = (RELU ? relu(X) : X)[N,128] @ W[128,128]
// One wave per 16x32 output tile (two 16x16 WMMA accumulators): the A
// fragment is loaded once per k-step and feeds two v_wmma_f32_16x16x4_f32,
// halving A-operand traffic. 100000 % 16 == 0 so tiling is exact and all
// branches are wave-uniform (EXEC all-ones at every WMMA).
template <bool RELU>
__global__ void k_gemm(const float* __restrict__ X, const float* __restrict__ W,
                       float* __restrict__ H) {
  const int lane  = threadIdx.x & 31;
  const int wave  = blockIdx.x * (blockDim.x >> 5) + (threadIdx.x >> 5);
  const int tm    = wave / TILES_N2;           // row tile (wave-uniform)
  const int tn    = wave % TILES_N2;           // 32-wide col tile (wave-uniform)
  if (tm * 16 >= N_NODES) return;              // wave-uniform guard

  const int ml    = lane & 15;                 // M (A) / N (B,C) within tile
  const int khalf = (lane >> 4) << 1;          // 0 or 2: K sub-pair per lane half
  const int n0    = tn * 32 + ml;

  // A frag: lane holds X[tm*16+ml][k+khalf+{0,1}]
  const float* aBase = X + (size_t)(tm * 16 + ml) * DIM + khalf;
  // B frags: lane holds W[k+khalf+{0,1}][n0] and [n0+16]
  const float* bBase = W + (size_t)khalf * DIM + n0;

  v8f c0 = {}, c1 = {};
#pragma unroll
  for (int k = 0; k < DIM; k += 4) {
    float a0 = aBase[k + 0];
    float a1 = aBase[k + 1];
    if (RELU) { a0 = fmaxf(a0, 0.0f); a1 = fmaxf(a1, 0.0f); }
    v2f a, b0, b1;
    a[0]  = a0;
    a[1]  = a1;
    b0[0] = bBase[(size_t)k * DIM];
    b0[1] = bBase[(size_t)k * DIM + DIM];
    b1[0] = bBase[(size_t)k * DIM + 16];
    b1[1] = bBase[(size_t)k * DIM + DIM + 16];
    // 8 args: (neg_a, A, neg_b, B, c_mod, C, reuse_a, reuse_b)
    c0 = __builtin_amdgcn_wmma_f32_16x16x4_f32(false, a, false, b0,
                                               (short)0, c0, false, false);
    c1 = __builtin_amdgcn_wmma_f32_16x16x4_f32(false, a, false, b1,
                                               (short)0, c1, false, false);
  }

  // C/D layout: VGPR r -> (M = r + 8*(lane>=16), N = lane&15)
  float* out = H + (size_t)(tm * 16 + (lane >> 4) * 8) * DIM + n0;
#pragma unroll
  for (int r = 0; r < 8; ++r) {
    out[(size_t)r * DIM]      = c0[r];
    out[(size_t)r * DIM + 16] = c1[r];
  }
}

// ---------------- accumulator init: bias + self-loop ----------------
// ACC[i][f] = bias[f] + dinv[i]^2 * H[i][f]
__global__ void k_init_self(const float* __restrict__ H, const float* __restrict__ dinv,
                            const float* __restrict__ bias, float* __restrict__ ACC) {
  long long idx = (long long)blockIdx.x * blockDim.x + threadIdx.x;
  if (idx < (long long)N_NODES * DIM) {
    int node = (int)(idx >> 7);
    int f    = (int)(idx & (DIM - 1));
    float w  = dinv[node];
    ACC[idx] = bias[f] + w * w * H[idx];
  }
}

// ---------------- edge scatter: ACC[col] += norm * H[row] ----------------
// One wave per edge; 128 features -> float4 per lane, 4 f32 atomics per lane.
__global__ void k_scatter(const int* __restrict__ row, const int* __restrict__ col,
                          const float* __restrict__ dinv, const float* __restrict__ H,
                          float* __restrict__ ACC) {
  const int lane = threadIdx.x & 31;
  const int e    = blockIdx.x * (blockDim.x >> 5) + (threadIdx.x >> 5);
  if (e >= N_EDGES) return;
  const int s = row[e];
  const int d = col[e];
  const float nrm = dinv[s] * dinv[d];
  const float4 v = ((const float4*)(H + (size_t)s * DIM))[lane];
  float* a = ACC + (size_t)d * DIM + lane * 4;
  atomicAdd(a + 0, nrm * v.x);
  atomicAdd(a + 1, nrm * v.y);
  atomicAdd(a + 2, nrm * v.z);
  atomicAdd(a + 3, nrm * v.w);
}

// ---------------- launch ----------------

extern "C" void kernel_launch(void* const* d_in, const int* in_sizes, int n_in,
                              void* d_out, int out_size, void* d_ws, size_t ws_size,
                              hipStream_t stream) {
  const float* x   = (const float*)d_in[0];
  const int*   ei  = (const int*)d_in[1];       // edge_index [2, E]
  const float* W0  = (const float*)d_in[2];
  const float* b0  = (const float*)d_in[3];
  const float* W1  = (const float*)d_in[4];
  const float* b1  = (const float*)d_in[5];
  const float* W2  = (const float*)d_in[6];
  const float* b2  = (const float*)d_in[7];
  float*       out = (float*)d_out;

  const int* row = ei;            // sources
  const int* col = ei + N_EDGES;  // destinations

  // workspace layout: dinv | H | ACC
  float* dinv = (float*)d_ws;
  float* H    = dinv + ((N_NODES + 255) & ~255);
  float* ACC  = H + (size_t)N_NODES * DIM;

  const int T = 256;
  const int gN    = (N_NODES + T - 1) / T;
  const int gE    = (N_EDGES + T - 1) / T;
  const int gNF   = ((long long)N_NODES * DIM + T - 1) / T;
  const int gGemm = (N_NODES / 16) * TILES_N2 / (T / 32);  // 3125 blocks, 8 waves each
  const int gScat = (N_EDGES + (T / 32) - 1) / (T / 32);   // wave per edge

  // normalization: deg -> dinv (in place)
  k_deg_init <<<gN, T, 0, stream>>>(dinv);
  k_deg_count<<<gE, T, 0, stream>>>(col, dinv);
  k_dinv     <<<gN, T, 0, stream>>>(dinv);

  // layer 0
  k_gemm<false><<<gGemm, T, 0, stream>>>(x, W0, H);
  k_init_self  <<<gNF, T, 0, stream>>>(H, dinv, b0, ACC);
  k_scatter    <<<gScat, T, 0, stream>>>(row, col, dinv, H, ACC);

  // layer 1 (relu fused into GEMM A-read)
  k_gemm<true> <<<gGemm, T, 0, stream>>>(ACC, W1, H);
  k_init_self  <<<gNF, T, 0, stream>>>(H, dinv, b1, ACC);
  k_scatter    <<<gScat, T, 0, stream>>>(row, col, dinv, H, ACC);

  // layer 2 -> d_out (no trailing relu)
  k_gemm<true> <<<gGemm, T, 0, stream>>>(ACC, W2, H);
  k_init_self  <<<gNF, T, 0, stream>>>(H, dinv, b2, out);
  k_scatter    <<<gScat, T, 0, stream>>>(row, col, dinv, H, out);
}